// neural_network_34763465293890
// MI455X (gfx1250) — compile-verified
//
#include <hip/hip_runtime.h>
#include <math.h>

// ---- geometry ----
#define T_TOTAL 1024
#define FEAT    16
#define LSZ     256          // material layer width
#define BULK    64
#define COH     32
#define TILE    16           // timesteps per WMMA tile
#define NWAVE   4
#define NTHREADS 128

// ---- material constants (match reference) ----
#define E_MOD   200.0f
#define NU      0.3f
#define SY0     2.0f
#define HARD    10.0f
#define G_MOD   (E_MOD / (2.0f * (1.0f + NU)))          // 76.923077
#define C11     (E_MOD / (1.0f - NU * NU))              // 219.78022
#define C12     (C11 * NU)
#define C33     (C11 * (1.0f - NU) * 0.5f)              // == G_MOD
#define K_PEN   50.0f
#define DELTA0  0.1f
#define DELTAF  1.0f

typedef __attribute__((ext_vector_type(2))) float v2f;
typedef __attribute__((ext_vector_type(8))) float v8f;
typedef __attribute__((ext_vector_type(2))) int   i32x2;

// CDNA5 async global<->LDS path (ASYNCcnt). Guarded so the kernel still
// compiles (synchronous fallback) if the toolchain lacks these builtins.
#if __has_builtin(__builtin_amdgcn_global_load_async_to_lds_b64) && \
    __has_builtin(__builtin_amdgcn_global_store_async_from_lds_b64) && \
    __has_builtin(__builtin_amdgcn_s_wait_asynccnt)
#define USE_ASYNC 1
typedef i32x2 __attribute__((address_space(1)))* gp64;   // global v2i*
typedef i32x2 __attribute__((address_space(3)))* lp64;   // LDS    v2i*
#define ASYNC_LOAD64(gsrc, ldst)                                             \
    __builtin_amdgcn_global_load_async_to_lds_b64(                           \
        (gp64)(i32x2*)(void*)(gsrc), (lp64)(i32x2*)(void*)(ldst), 0, 0)
#define ASYNC_STORE64(gdst, lsrc)                                            \
    __builtin_amdgcn_global_store_async_from_lds_b64(                        \
        (gp64)(i32x2*)(void*)(gdst), (lp64)(i32x2*)(void*)(lsrc), 0, 0)
#else
#define USE_ASYNC 0
#endif

__device__ __forceinline__ float softplusf(float w) {
    return (w > 20.0f) ? w : log1pf(__expf(w));
}

__global__ __launch_bounds__(NTHREADS)
void fused_material_rnn(const float* __restrict__ x,     // [256,1024,16]
                        const float* __restrict__ W1,    // [256,16]
                        const float* __restrict__ W2,    // [16,256]
                        float* __restrict__ out)         // [256,1024,16]
{
    __shared__ float sW1[LSZ][FEAT];        // [l][f]                 16 KB
    __shared__ float sW2[LSZ][FEAT];        // [l][o]=sp(W2[o][l])    16 KB
    __shared__ float sX [2][TILE][FEAT];    // double-buffered x tile  2 KB
    __shared__ float sH [TILE][LSZ];        // h tile (y in place)    16 KB
    __shared__ float sP [NWAVE][TILE][TILE];// fc2 per-wave partials   4 KB
    __shared__ float sO [TILE][FEAT];       // out tile staging        1 KB

    const int b    = blockIdx.x;
    const int tid  = threadIdx.x;
    const int wave = tid >> 5;
    const int lane = tid & 31;
    const int ln   = lane & 15;          // M (A/C row) or N (B/C col) index
    const int kb   = (lane >> 4) << 1;   // K base for this half-wave (0 or 2)
    const int mhi  = (lane >> 4) << 3;   // C/D row base (0 or 8)

    // ---- stage weights (once per block) ----
    for (int i = tid; i < LSZ * FEAT; i += NTHREADS)
        sW1[i >> 4][i & 15] = W1[i];                       // W1 row-major [l][f]
    for (int i = tid; i < FEAT * LSZ; i += NTHREADS) {
        int o = i >> 8, l = i & 255;
        sW2[l][o] = softplusf(W2[i]);                      // W2 is [o][l]
    }

    // ---- per-thread material state (registers across all 1024 steps) ----
    const bool isBulk = (tid < BULK);
    const bool isCoh  = (tid >= BULK) && (tid < BULK + COH);
    float p0 = 0.f, p1 = 0.f, p2 = 0.f;   // plastic strain (bulk)
    float epbar = 0.f;                    // accumulated plastic strain (bulk)
    float hist  = 0.f;                    // damage history (cohesive)

    const float* xb = x   + (size_t)b * T_TOTAL * FEAT;
    float*       ob = out + (size_t)b * T_TOTAL * FEAT;

#if USE_ASYNC
    // pre-issue tile 0: each lane moves 8 bytes of the 1 KB tile
    ASYNC_LOAD64(xb + tid * 2, &sX[0][0][0] + tid * 2);
#endif
    __syncthreads();   // weights staged

    for (int t0 = 0; t0 < T_TOTAL; t0 += TILE) {
        const int buf = (t0 >> 4) & 1;

#if USE_ASYNC
        __builtin_amdgcn_s_wait_asynccnt(0);   // tile `buf` landed, out tile flushed
        __syncthreads();
        if (t0 + TILE < T_TOTAL) {             // kick off next tile's copy now;
            ASYNC_LOAD64(xb + (t0 + TILE) * FEAT + tid * 2,   // overlaps the
                         &sX[buf ^ 1][0][0] + tid * 2);       // scan + fc2 below
        }
#else
        for (int i = tid; i < TILE * FEAT; i += NTHREADS)
            (&sX[buf][0][0])[i] = xb[t0 * FEAT + i];
        __syncthreads();
#endif
        if (t0 + 2 * TILE < T_TOTAL)  // distance-2 prefetch -> global_prefetch_b8
            __builtin_prefetch(&xb[(t0 + 2 * TILE) * FEAT + tid * 2], 0, 1);

        // ---- fc1: H[16x256] = X[16x16] @ W1^T  (v_wmma_f32_16x16x4_f32) ----
        for (int c = wave * 4; c < wave * 4 + 4; ++c) {
            v8f acc = {0.f, 0.f, 0.f, 0.f, 0.f, 0.f, 0.f, 0.f};
#pragma unroll
            for (int kc = 0; kc < 4; ++kc) {
                v2f a, bb;
                a.x  = sX[buf][ln][kc * 4 + kb];
                a.y  = sX[buf][ln][kc * 4 + kb + 1];
                bb.x = sW1[c * 16 + ln][kc * 4 + kb];      // B[k][n] = W1[n][k]
                bb.y = sW1[c * 16 + ln][kc * 4 + kb + 1];
                acc = __builtin_amdgcn_wmma_f32_16x16x4_f32(
                        false, a, false, bb, (short)0, acc, false, false);
            }
#pragma unroll
            for (int r = 0; r < 8; ++r)
                sH[mhi + r][c * 16 + ln] = acc[r];
        }
        __syncthreads();

        // ---- sequential material scan inside the tile (state in registers) ----
        if (isBulk) {
            const int p = tid;
#pragma unroll
            for (int tt = 0; tt < TILE; ++tt) {
                float e0 = sH[tt][3 * p], e1 = sH[tt][3 * p + 1], e2 = sH[tt][3 * p + 2];
                float d0 = e0 - p0, d1 = e1 - p1, d2 = e2 - p2;
                float s1  = C11 * d0 + C12 * d1;
                float s2  = C12 * d0 + C11 * d1;
                float s12 = C33 * d2;
                float seq = sqrtf(fmaxf(s1 * s1 - s1 * s2 + s2 * s2 + 3.f * s12 * s12, 1e-12f));
                float fy  = seq - (SY0 + HARD * epbar);
                bool  pl  = fy > 0.f;
                float dep = pl ? fy * (1.f / (3.f * G_MOD + HARD)) : 0.f;
                float r   = (SY0 + HARD * (epbar + dep)) / seq;
                float o0  = pl ? r * s1  : s1;
                float o1  = pl ? r * s2  : s2;
                float o2  = pl ? r * s12 : s12;
                if (pl) {   // epsp_new = eps - C_INV @ sig
                    p0 = e0 - (o0 - NU * o1) * (1.f / E_MOD);
                    p1 = e1 - (o1 - NU * o0) * (1.f / E_MOD);
                    p2 = e2 - o2 * (1.f / G_MOD);
                }
                epbar += dep;
                sH[tt][3 * p]     = o0;
                sH[tt][3 * p + 1] = o1;
                sH[tt][3 * p + 2] = o2;
            }
        } else if (isCoh) {
            const int p = tid - BULK;
#pragma unroll
            for (int tt = 0; tt < TILE; ++tt) {
                float dn = sH[tt][BULK * 3 + 2 * p];
                float ds = sH[tt][BULK * 3 + 2 * p + 1];
                float dnp = fmaxf(dn, 0.f);
                float lam = sqrtf(fmaxf(dnp * dnp + ds * ds, 1e-12f));
                hist = fmaxf(hist, lam);
                float dmg = DELTAF * (hist - DELTA0) /
                            (fmaxf(hist, 1e-12f) * (DELTAF - DELTA0));
                dmg = fminf(fmaxf(dmg, 0.f), 1.f);
                float tn = K_PEN * dn * ((dn > 0.f) ? (1.f - dmg) : 1.f);
                float ts = (1.f - dmg) * K_PEN * ds;
                sH[tt][BULK * 3 + 2 * p]     = tn;
                sH[tt][BULK * 3 + 2 * p + 1] = ts;
            }
        }
        __syncthreads();

        // ---- fc2: Out[16x16] = Y[16x256] @ softplus(W2)^T ----
        {
            v8f acc = {0.f, 0.f, 0.f, 0.f, 0.f, 0.f, 0.f, 0.f};
            for (int kc = wave * 16; kc < wave * 16 + 16; ++kc) {
                v2f a, bb;
                a.x  = sH[ln][kc * 4 + kb];
                a.y  = sH[ln][kc * 4 + kb + 1];
                bb.x = sW2[kc * 4 + kb][ln];               // B[k][n] = sp(W2)[n][k]
                bb.y = sW2[kc * 4 + kb + 1][ln];
                acc = __builtin_amdgcn_wmma_f32_16x16x4_f32(
                        false, a, false, bb, (short)0, acc, false, false);
            }
#pragma unroll
            for (int r = 0; r < 8; ++r)
                sP[wave][mhi + r][ln] = acc[r];
        }
        __syncthreads();

        // ---- cross-wave reduction, then push out tile asynchronously ----
        for (int i = tid; i < TILE * FEAT; i += NTHREADS) {
            int mm = i >> 4, nn = i & 15;
            sO[mm][nn] = sP[0][mm][nn] + sP[1][mm][nn] +
                         sP[2][mm][nn] + sP[3][mm][nn];
        }
        __syncthreads();
#if USE_ASYNC
        // LDS -> global, tracked by ASYNCcnt; waited at next iteration's top
        ASYNC_STORE64(ob + t0 * FEAT + tid * 2, &sO[0][0] + tid * 2);
#else
        {
            float2 v = *(const float2*)(&sO[0][0] + tid * 2);
            *(float2*)(ob + t0 * FEAT + tid * 2) = v;
        }
        __syncthreads();   // sP/sO reuse protection
#endif
    }
#if USE_ASYNC
    __builtin_amdgcn_s_wait_asynccnt(0);   // drain final out-tile store
#endif
}

extern "C" void kernel_launch(void* const* d_in, const int* in_sizes, int n_in,
                              void* d_out, int out_size, void* d_ws, size_t ws_size,
                              hipStream_t stream) {
    const float* x  = (const float*)d_in[0];   // [256,1024,16]
    const float* W1 = (const float*)d_in[1];   // [256,16]
    const float* W2 = (const float*)d_in[2];   // [16,256]
    float* out = (float*)d_out;                // [256,1024,16]
    (void)in_sizes; (void)n_in; (void)out_size; (void)d_ws; (void)ws_size;
    fused_material_rnn<<<256, NTHREADS, 0, stream>>>(x, W1, W2, out);
}